// Model_85641647882346
// MI455X (gfx1250) — compile-verified
//
#include <hip/hip_runtime.h>
#include <math.h>

#define N_USER 50000
#define N_REPO 100000
#define NEDGE  1000000
#define NEP    200000
#define D_IN   256
#define D_EMB  125
#define LD_EMB 128
#define D_HID  96
#define D_OUT  64
#define EPSN   1e-12f
#define MAX_K  256

typedef float v2f __attribute__((ext_vector_type(2)));
typedef float v8f __attribute__((ext_vector_type(8)));

// ---------------------------------------------------------------------------
// Generic fp32 WMMA GEMM:  C = epi( (A ⊙ rsqrt(max(degA,1))) @ W + bias
//                                    + agg ⊙ rsqrt(max(degC,1)) + bias2 )
// One wave -> one 16x16 C tile via V_WMMA_F32_16X16X4_F32, K stepped by 4.
// W n-tile is staged transposed in LDS (shared by the block's 4 waves) with
// zero padding for n >= Nreal and k >= K, so the K-loop has NO guards and
// NO divergence (EXEC stays all-ones as WMMA requires).
// Caller guarantees A columns [K, Kpad) are readable and zero (only K=125
// needs this; its A operands are stored ld=128 with zeroed pad columns).
// A-frag: lanes 0-15 hold {k,k+1} of row m0+lane; lanes 16-31 hold {k+2,k+3}.
// B-frag: lanes 0-15 hold col n0+lane of rows {k,k+1}; lanes 16-31 {k+2,k+3}.
// C/D: VGPR v, lanes 0-15 -> (m0+v, n0+lane); lanes 16-31 -> (m0+8+v, ...).
// ---------------------------------------------------------------------------
__global__ void __launch_bounds__(128)
gemm_wmma_f32(const float* __restrict__ A, int lda,
              const float* __restrict__ W, int ldw,
              const float* __restrict__ degA,   // nullable: row pre-scale (raw degree)
              const float* __restrict__ bias,   // nullable: dense bias [Nreal]
              const float* __restrict__ agg,    // nullable: M x ldc aggregation
              const float* __restrict__ degC,   // with agg: in-degree (raw)
              const float* __restrict__ bias2,  // with agg: conv bias [Nreal]
              float* __restrict__ C, int ldc,
              int M, int Nreal, int K, int Kpad, int relu)
{
  __shared__ __align__(16) float Wt[16 * (MAX_K + 4)];
  const int ldsStride = Kpad + 4;          // de-bank the 16 lanes
  const int n0 = blockIdx.x * 16;

  // ---- cooperative staging: Wt[n][k] = W[k][n0+n], zero-padded ----
  {
    const int tid = threadIdx.y * 32 + threadIdx.x;
    const int sn  = tid >> 3;              // 0..15
    const int sk0 = tid & 7;               // 0..7
    const int gn  = n0 + sn;
    const bool gok = (gn < Nreal);
    float* dstp = &Wt[sn * ldsStride];
    for (int k = sk0; k < Kpad; k += 8)
      dstp[k] = (gok && k < K) ? W[(long)k * ldw + gn] : 0.0f;
  }
  __syncthreads();

  const int lane = threadIdx.x;
  const int half = lane >> 4;              // 0: lanes 0-15, 1: lanes 16-31
  const int l16  = lane & 15;
  const int mtile = blockIdx.y * blockDim.y + threadIdx.y;
  if (mtile * 16 >= M) return;             // uniform per wave; after the barrier
  const int m0 = mtile * 16;

  const int   mA = m0 + l16;
  const float sA = degA ? rsqrtf(fmaxf(degA[mA], 1.0f)) : 1.0f;

  const float* Ap = A + (long)mA * lda + 2 * half;   // this lane's first K-pair
  const float* Wp = &Wt[l16 * ldsStride + 2 * half];

  v8f acc = {};
#pragma unroll 4
  for (int k0 = 0; k0 < Kpad; k0 += 4) {
    v2f a = *(const v2f*)(Ap + k0);        // global_load_b64, unguarded
    a *= sA;
    v2f b = *(const v2f*)(Wp + k0);        // ds_load_b64, unguarded
    acc = __builtin_amdgcn_wmma_f32_16x16x4_f32(false, a, false, b,
                                                (short)0, acc, false, false);
  }

  const int   n   = n0 + l16;
  const float bn  = (bias  && n < Nreal) ? bias[n]  : 0.0f;
  const float b2n = (bias2 && n < Nreal) ? bias2[n] : 0.0f;
#pragma unroll
  for (int v = 0; v < 8; ++v) {
    const int m = m0 + v + 8 * half;
    float val = acc[v] + bn;
    if (agg)  val += agg[(long)m * ldc + n] * rsqrtf(fmaxf(degC[m], 1.0f)) + b2n;
    if (relu) val = fmaxf(val, 0.0f);
    C[(long)m * ldc + n] = (n < Nreal) ? val : 0.0f;   // zero-fill pad columns
  }
}

// degrees: segment-count of src and dst indices
__global__ void degree_kernel(const int* __restrict__ src, const int* __restrict__ dst,
                              float* __restrict__ dsrc, float* __restrict__ ddst, int E)
{
  int e = blockIdx.x * blockDim.x + threadIdx.x;
  if (e < E) {
    atomicAdd(&dsrc[src[e]], 1.0f);
    atomicAdd(&ddst[dst[e]], 1.0f);
  }
}

// edge scatter-add: agg[dst[e], :] += feat[src[e], :]   (blockDim.x == D)
__global__ void scatter_add_kernel(const float* __restrict__ feat,
                                   const int* __restrict__ src,
                                   const int* __restrict__ dst,
                                   float* __restrict__ agg, int E, int D)
{
  int e = blockIdx.x * blockDim.y + threadIdx.y;
  if (e >= E) return;
  int s = src[e], d = dst[e];
  atomicAdd(&agg[(long)d * D + threadIdx.x], feat[(long)s * D + threadIdx.x]);
}

// inverse L2 norm per row of a rows x 64 matrix (one wave per row)
__global__ void rownorm_kernel(const float* __restrict__ X, float* __restrict__ invn, int rows)
{
  int r = blockIdx.x * blockDim.y + threadIdx.y;
  if (r >= rows) return;
  int lane = threadIdx.x;
  float2 v = ((const float2*)(X + (long)r * D_OUT))[lane];
  float s = v.x * v.x + v.y * v.y;
#pragma unroll
  for (int o = 16; o > 0; o >>= 1) s += __shfl_xor(s, o, 32);
  if (lane == 0) invn[r] = 1.0f / fmaxf(sqrtf(s), EPSN);
}

// cosine score per edge pair (one wave per edge)
__global__ void score_kernel(const float* __restrict__ HU, const float* __restrict__ HR,
                             const float* __restrict__ invU, const float* __restrict__ invR,
                             const int* __restrict__ iu, const int* __restrict__ ir,
                             float* __restrict__ out, int Ep)
{
  int e = blockIdx.x * blockDim.y + threadIdx.y;
  if (e >= Ep) return;
  int lane = threadIdx.x;
  int u = iu[e], r = ir[e];
  float2 a = ((const float2*)(HU + (long)u * D_OUT))[lane];
  float2 b = ((const float2*)(HR + (long)r * D_OUT))[lane];
  float s = a.x * b.x + a.y * b.y;
#pragma unroll
  for (int o = 16; o > 0; o >>= 1) s += __shfl_xor(s, o, 32);
  if (lane == 0) out[e] = s * invU[u] * invR[r];
}

extern "C" void kernel_launch(void* const* d_in, const int* in_sizes, int n_in,
                              void* d_out, int out_size, void* d_ws, size_t ws_size,
                              hipStream_t stream)
{
  const float* user_feat = (const float*)d_in[0];
  const float* repo_feat = (const float*)d_in[1];
  const float* W_ue = (const float*)d_in[2];   const float* b_ue = (const float*)d_in[3];
  const float* W_re = (const float*)d_in[4];   const float* b_re = (const float*)d_in[5];
  const float* W_hid_ur = (const float*)d_in[6];  const float* b_hid_ur = (const float*)d_in[7];
  const float* W_hid_ru = (const float*)d_in[8];  const float* b_hid_ru = (const float*)d_in[9];
  const float* W_out_ur = (const float*)d_in[10]; const float* b_out_ur = (const float*)d_in[11];
  const float* W_out_ru = (const float*)d_in[12]; const float* b_out_ru = (const float*)d_in[13];
  const float* W_hcu = (const float*)d_in[14]; const float* b_hcu = (const float*)d_in[15];
  const float* W_hcr = (const float*)d_in[16]; const float* b_hcr = (const float*)d_in[17];
  const float* W_ocu = (const float*)d_in[18]; const float* b_ocu = (const float*)d_in[19];
  const float* W_ocr = (const float*)d_in[20]; const float* b_ocr = (const float*)d_in[21];
  const int* e_ur_src = (const int*)d_in[22];  const int* e_ur_dst = (const int*)d_in[23];
  const int* e_ru_src = (const int*)d_in[24];  const int* e_ru_dst = (const int*)d_in[25];
  const int* pos_u = (const int*)d_in[26];     const int* pos_r = (const int*)d_in[27];
  const int* neg_u = (const int*)d_in[28];     const int* neg_r = (const int*)d_in[29];
  float* out = (float*)d_out;

  // -------- workspace carve-up (floats) --------
  float* w = (float*)d_ws;
  size_t o = 0;
  float* deg_u_out_ur = w + o; o += N_USER;   // segsum over e_ur_src
  float* deg_r_in_ur  = w + o; o += N_REPO;   // segsum over e_ur_dst
  float* deg_r_out_ru = w + o; o += N_REPO;   // segsum over e_ru_src
  float* deg_u_in_ru  = w + o; o += N_USER;   // segsum over e_ru_dst
  float* agg_u   = w + o; o += (size_t)N_USER * D_HID;   // \ contiguous for one memset
  float* agg_r   = w + o; o += (size_t)N_REPO * D_HID;   // /
  float* h_user  = w + o; o += (size_t)N_USER * LD_EMB;  // reused as h_user_new (ld 64)
  float* h_repo  = w + o; o += (size_t)N_REPO * LD_EMB;  // reused as h_repo_new (ld 64)
  float* feat_u  = w + o; o += (size_t)N_USER * D_HID;   // reused layer-2 (ld 64)
  float* feat_r  = w + o; o += (size_t)N_REPO * D_HID;
  float* out_user = w + o; o += (size_t)N_USER * D_HID;
  float* out_repo = w + o; o += (size_t)N_REPO * D_HID;
  float* inv_nu  = w + o; o += N_USER;
  float* inv_nr  = w + o; o += N_REPO;

  const size_t degBytes = (size_t)(2 * N_USER + 2 * N_REPO) * sizeof(float);
  const size_t aggBytes = ((size_t)N_USER + N_REPO) * D_HID * sizeof(float);

  auto gemm = [&](const float* A, int lda, const float* W_, int ldw,
                  const float* degA, const float* bias,
                  const float* agg, const float* degC, const float* bias2,
                  float* C, int ldc, int M, int Nreal, int K, int relu) {
    const int Kpad = (K + 3) & ~3;
    dim3 blk(32, 4);
    dim3 grd(ldc / 16, (M / 16 + 3) / 4);
    hipLaunchKernelGGL(gemm_wmma_f32, grd, blk, 0, stream,
                       A, lda, W_, ldw, degA, bias, agg, degC, bias2,
                       C, ldc, M, Nreal, K, Kpad, relu);
  };

  auto scatter = [&](const float* feat, const int* src, const int* dst,
                     float* agg, int D) {
    const int epb = 256 / D;               // edges per block (2 for 96, 4 for 64)
    dim3 blk(D, epb);
    dim3 grd((NEDGE + epb - 1) / epb);
    hipLaunchKernelGGL(scatter_add_kernel, grd, blk, 0, stream,
                       feat, src, dst, agg, NEDGE, D);
  };

  // ---- degrees (shared by both layers) ----
  hipMemsetAsync(deg_u_out_ur, 0, degBytes, stream);
  hipMemsetAsync(agg_u, 0, aggBytes, stream);
  hipLaunchKernelGGL(degree_kernel, dim3((NEDGE + 255) / 256), dim3(256), 0, stream,
                     e_ur_src, e_ur_dst, deg_u_out_ur, deg_r_in_ur, NEDGE);
  hipLaunchKernelGGL(degree_kernel, dim3((NEDGE + 255) / 256), dim3(256), 0, stream,
                     e_ru_src, e_ru_dst, deg_r_out_ru, deg_u_in_ru, NEDGE);

  // ---- embeddings: h = feat @ W + b  (256 -> 125, stored ld=128, pads zeroed) ----
  gemm(user_feat, D_IN, W_ue, D_EMB, nullptr, b_ue, nullptr, nullptr, nullptr,
       h_user, LD_EMB, N_USER, D_EMB, D_IN, 0);
  gemm(repo_feat, D_IN, W_re, D_EMB, nullptr, b_re, nullptr, nullptr, nullptr,
       h_repo, LD_EMB, N_REPO, D_EMB, D_IN, 0);

  // ---- layer 1 conv features: (h ⊙ rsqrt(out-deg)) @ W_hid  (125 -> 96) ----
  // K=125 -> Kpad=128: h_* pad columns are zero, so unguarded reads are exact.
  gemm(h_user, LD_EMB, W_hid_ur, D_HID, deg_u_out_ur, nullptr, nullptr, nullptr, nullptr,
       feat_u, D_HID, N_USER, D_HID, D_EMB, 0);
  gemm(h_repo, LD_EMB, W_hid_ru, D_HID, deg_r_out_ru, nullptr, nullptr, nullptr, nullptr,
       feat_r, D_HID, N_REPO, D_HID, D_EMB, 0);

  // ---- layer 1 edge scatter (SpMM) ----
  scatter(feat_u, e_ur_src, e_ur_dst, agg_r, D_HID);
  scatter(feat_r, e_ru_src, e_ru_dst, agg_u, D_HID);

  // ---- layer 1 combine: relu(h @ W_self + b_self + agg*rsqrt(in-deg) + b_conv) ----
  gemm(h_user, LD_EMB, W_hcu, D_HID, nullptr, b_hcu, agg_u, deg_u_in_ru, b_hid_ru,
       out_user, D_HID, N_USER, D_HID, D_EMB, 1);
  gemm(h_repo, LD_EMB, W_hcr, D_HID, nullptr, b_hcr, agg_r, deg_r_in_ur, b_hid_ur,
       out_repo, D_HID, N_REPO, D_HID, D_EMB, 1);

  // ---- layer 2 ----
  hipMemsetAsync(agg_u, 0, aggBytes, stream);   // covers the D_OUT-strided use too
  gemm(out_user, D_HID, W_out_ur, D_OUT, deg_u_out_ur, nullptr, nullptr, nullptr, nullptr,
       feat_u, D_OUT, N_USER, D_OUT, D_HID, 0);
  gemm(out_repo, D_HID, W_out_ru, D_OUT, deg_r_out_ru, nullptr, nullptr, nullptr, nullptr,
       feat_r, D_OUT, N_REPO, D_OUT, D_HID, 0);

  scatter(feat_u, e_ur_src, e_ur_dst, agg_r, D_OUT);
  scatter(feat_r, e_ru_src, e_ru_dst, agg_u, D_OUT);

  gemm(out_user, D_HID, W_ocu, D_OUT, nullptr, b_ocu, agg_u, deg_u_in_ru, b_out_ru,
       h_user, D_OUT, N_USER, D_OUT, D_HID, 1);   // h_user_new (ld 64)
  gemm(out_repo, D_HID, W_ocr, D_OUT, nullptr, b_ocr, agg_r, deg_r_in_ur, b_out_ur,
       h_repo, D_OUT, N_REPO, D_OUT, D_HID, 1);   // h_repo_new (ld 64)

  // ---- cosine scores ----
  hipLaunchKernelGGL(rownorm_kernel, dim3((N_USER + 7) / 8), dim3(32, 8), 0, stream,
                     h_user, inv_nu, N_USER);
  hipLaunchKernelGGL(rownorm_kernel, dim3((N_REPO + 7) / 8), dim3(32, 8), 0, stream,
                     h_repo, inv_nr, N_REPO);
  hipLaunchKernelGGL(score_kernel, dim3((NEP + 7) / 8), dim3(32, 8), 0, stream,
                     h_user, h_repo, inv_nu, inv_nr, pos_u, pos_r, out, NEP);
  hipLaunchKernelGGL(score_kernel, dim3((NEP + 7) / 8), dim3(32, 8), 0, stream,
                     h_user, h_repo, inv_nu, inv_nr, neg_u, neg_r, out + NEP, NEP);
}